// TSRL_69604239999118
// MI455X (gfx1250) — compile-verified
//
#include <hip/hip_runtime.h>
#include <math.h>

// Problem constants (match reference)
#define NV    35      // views
#define VD    3       // view dim
#define C1    64
#define C2    128
#define EMB   128
#define BB    64      // batch
#define SS    1024    // seq
#define L1    512     // after maxpool2
#define EPSB  1e-5f

typedef __attribute__((ext_vector_type(2))) float v2f;
typedef __attribute__((ext_vector_type(8))) float v8f;
typedef __attribute__((ext_vector_type(4))) int   v4i;

// gfx1250 async global->LDS copy (ASYNCcnt-tracked), guarded for portability.
#ifdef __has_builtin
#if __has_builtin(__builtin_amdgcn_global_load_async_to_lds_b128) && \
    __has_builtin(__builtin_amdgcn_s_wait_asynccnt)
#define USE_ASYNC_LDS 1
#endif
#endif
#ifndef USE_ASYNC_LDS
#define USE_ASYNC_LDS 0
#endif

__device__ __forceinline__ float gelu_exact(float x) {
    return 0.5f * x * (1.0f + erff(x * 0.70710678118654752440f));
}

// LDS layout (floats) for the fused kernel
#define OFF_XBUF 0                      // 138
#define OFF_HBUF 138                    // 3*136 = 408
#define OFF_W1   546                    // 960
#define OFF_S1   1506                   // 64
#define OFF_A1   1570                   // 64
#define OFF_S2   1634                   // 128
#define OFF_A2   1762                   // 128
#define OFF_H1T  1890                   // 66*68 = 4488  (transposed, pitch 68)
#define OFF_W2   6380                   // 128*196 = 25088 (pitch 196; 16B-aligned rows)
#define SMEM_FLOATS (OFF_W2 + 128 * 196)   // 31468 floats = 125872 B

// ---------------------------------------------------------------------------
// Kernel 0: zero the feats accumulator (ws) — graph-replay safe.
// ---------------------------------------------------------------------------
__global__ void tsrl_zero(float* __restrict__ p, int n) {
    int i = blockIdx.x * blockDim.x + threadIdx.x;
    if (i < n) p[i] = 0.0f;
}

// ---------------------------------------------------------------------------
// Kernel 1: fully fused per-view CNN.
//   grid = NV * BB * 8 tiles (64 output positions each), block = 256 (8 waves)
//   async  : c2_w -> LDS via GLOBAL_LOAD_ASYNC_TO_LDS_B128 (overlaps A+B)
//   stage A: grouped conv(1->3,k3,p1)+BN0+GELU  -> LDS h[3][136]
//   stage B: conv1(3->64,k5,p2)+BN1+GELU+pool2  -> LDS h1t[66][64] (transposed)
//   stage C: conv2(64->128,k3,p1) as 3 shifted GEMMs on V_WMMA_F32_16X16X4_F32
//            + BN2 + GELU + partial mean -> atomicAdd feats[v][b][128]
// ---------------------------------------------------------------------------
__global__ __launch_bounds__(256)
void tsrl_fused(const float* __restrict__ x,
                const float* __restrict__ vc_w, const float* __restrict__ vc_b,
                const float* __restrict__ bn0_g, const float* __restrict__ bn0_b,
                const float* __restrict__ bn0_m, const float* __restrict__ bn0_v,
                const float* __restrict__ c1_w,  const float* __restrict__ c1_b,
                const float* __restrict__ bn1_g, const float* __restrict__ bn1_b,
                const float* __restrict__ bn1_m, const float* __restrict__ bn1_v,
                const float* __restrict__ c2_w,  const float* __restrict__ c2_b,
                const float* __restrict__ bn2_g, const float* __restrict__ bn2_b,
                const float* __restrict__ bn2_m, const float* __restrict__ bn2_v,
                float* __restrict__ feats)
{
    extern __shared__ float smem[];
    float* xbuf = smem + OFF_XBUF;
    float* hbuf = smem + OFF_HBUF;
    float* w1   = smem + OFF_W1;
    float* s1a  = smem + OFF_S1;
    float* a1a  = smem + OFF_A1;
    float* s2a  = smem + OFF_S2;
    float* a2a  = smem + OFF_A2;
    float* h1t  = smem + OFF_H1T;    // [lloc][ci], pitch 68
    float* w2   = smem + OFF_W2;     // [co][K],   pitch 196

    const int tid = threadIdx.x;
    const int blk = blockIdx.x;
    const int v   = blk / (BB * 8);
    const int rem = blk % (BB * 8);
    const int b   = rem / 8;
    const int t   = rem % 8;
    const int l0  = t * 64;          // output-position tile [l0, l0+64)
    const int s_lo = 2 * l0 - 4;     // first h position needed (conv1 halo)

    // ---- issue async conv2-weight fill first: overlaps stages A and B ----
#if USE_ASYNC_LDS
    {
        const float* wsrc = c2_w + (size_t)v * (C2 * 192);
        for (int i = tid; i < C2 * 48; i += 256) {       // 48 x b128 per row
            int co = i / 48, q = i - co * 48;
            __builtin_amdgcn_global_load_async_to_lds_b128(
                (v4i*)(wsrc + co * 192 + q * 4),
                (v4i*)(w2 + co * 196 + q * 4), 0, 0);
        }
    }
#else
    for (int i = tid; i < C2 * 192; i += 256) {
        int co = i / 192, kk = i - co * 192;
        w2[co * 196 + kk] = c2_w[(size_t)v * (C2 * 192) + i];
    }
#endif

    // ---- cooperative loads ----
    for (int j = tid; j < 138; j += 256) {
        int s = s_lo - 1 + j;        // x positions needed: [s_lo-1, s_lo+136]
        xbuf[j] = (s >= 0 && s < SS) ? x[((size_t)b * SS + s) * NV + v] : 0.0f;
    }
    if (tid < C1) {
        int c = v * C1 + tid;
        float s = bn1_g[c] * rsqrtf(bn1_v[c] + EPSB);
        s1a[tid] = s;
        a1a[tid] = c1_b[c] * s + bn1_b[c] - bn1_m[c] * s;  // folded bias+BN
    }
    if (tid < C2) {
        int c = v * C2 + tid;
        float s = bn2_g[c] * rsqrtf(bn2_v[c] + EPSB);
        s2a[tid] = s;
        a2a[tid] = c2_b[c] * s + bn2_b[c] - bn2_m[c] * s;
    }
    for (int i = tid; i < C1 * VD * 5; i += 256)
        w1[i] = c1_w[(size_t)v * (C1 * VD * 5) + i];
    __syncthreads();

    // ---- stage A: grouped conv + BN0 + GELU ----
    for (int i = tid; i < VD * 136; i += 256) {
        int d = i / 136, sl = i - d * 136;
        int s = s_lo + sl;
        float hv = 0.0f;
        if (s >= 0 && s < SS) {
            int c = v * VD + d;
            float gc = xbuf[sl] * vc_w[c * 3 + 0]
                     + xbuf[sl + 1] * vc_w[c * 3 + 1]
                     + xbuf[sl + 2] * vc_w[c * 3 + 2];
            float sc = bn0_g[c] * rsqrtf(bn0_v[c] + EPSB);
            float ad = vc_b[c] * sc + bn0_b[c] - bn0_m[c] * sc;
            hv = gelu_exact(gc * sc + ad);
        }
        hbuf[d * 136 + sl] = hv;   // 0 outside [0,SS): conv1's zero pad
    }
    __syncthreads();

    // ---- stage B: conv1 + BN1 + GELU + maxpool2 -> h1t[lloc][ci], lloc in [0,66) ----
    for (int i = tid; i < C1 * 66; i += 256) {
        int ci = i & 63, lloc = i >> 6;
        int l = l0 - 1 + lloc;                // pooled position (halo for conv2)
        float r = 0.0f;
        if (l >= 0 && l < L1) {
            const float* wr = &w1[ci * 15];
            int base = 2 * l - s_lo;          // hbuf index of s=2l
            float acc0 = 0.0f, acc1 = 0.0f;
#pragma unroll
            for (int d = 0; d < 3; ++d) {
                const float* hb = &hbuf[d * 136 + base - 2];
#pragma unroll
                for (int k5 = 0; k5 < 5; ++k5) {
                    float w = wr[d * 5 + k5];
                    acc0 += w * hb[k5];
                    acc1 += w * hb[k5 + 1];
                }
            }
            float s = s1a[ci], a = a1a[ci];
            r = fmaxf(gelu_exact(acc0 * s + a), gelu_exact(acc1 * s + a));
        }
        h1t[lloc * 68 + ci] = r;              // 0 outside [0,L1): conv2's zero pad
    }
    // drain the async weight fill, then make everything visible block-wide
#if USE_ASYNC_LDS
    __builtin_amdgcn_s_wait_asynccnt(0);
#endif
    __syncthreads();

    // ---- stage C: conv2 as 3 shifted GEMMs on fp32 WMMA ----
    // wave w: output rows [16w,16w+16); 4 N-tiles of 16 cover the 64 positions.
    const int wave = tid >> 5;
    const int lane = tid & 31;
    const int mrow = (wave << 4) + (lane & 15);
    const int koff = (lane >> 4) << 1;        // 0 (lanes 0-15) / 2 (lanes 16-31)
    const int ncol = lane & 15;

    v8f acc0 = {}, acc1 = {}, acc2 = {}, acc3 = {};
    const float* Wrow = &w2[mrow * 196];

#pragma unroll
    for (int k = 0; k < 3; ++k) {             // conv tap (shifted-GEMM trick)
        for (int cc = 0; cc < C1; cc += 4) {  // K-dim in steps of 4
            // A frag: 16x4 fp32 — lanes 0-15 hold K=0/1, lanes 16-31 K=2/3
            int p = (cc + koff) * 3 + k;
            v2f a;
            a.x = Wrow[p];
            a.y = Wrow[p + 3];
            // B frags: transposed layout -> each frag is one aligned b64 load
            const float* hrow = &h1t[(ncol + k) * 68 + cc + koff];
            v2f b0 = *(const v2f*)(hrow);
            v2f b1 = *(const v2f*)(hrow + 16 * 68);
            v2f b2 = *(const v2f*)(hrow + 32 * 68);
            v2f b3 = *(const v2f*)(hrow + 48 * 68);
            acc0 = __builtin_amdgcn_wmma_f32_16x16x4_f32(false, a, false, b0, (short)0, acc0, false, false);
            acc1 = __builtin_amdgcn_wmma_f32_16x16x4_f32(false, a, false, b1, (short)0, acc1, false, false);
            acc2 = __builtin_amdgcn_wmma_f32_16x16x4_f32(false, a, false, b2, (short)0, acc2, false, false);
            acc3 = __builtin_amdgcn_wmma_f32_16x16x4_f32(false, a, false, b3, (short)0, acc3, false, false);
        }
    }

    // ---- BN2 + GELU + partial mean, reduce over the 16 columns per half-wave ----
#pragma unroll
    for (int j = 0; j < 8; ++j) {
        int co = (wave << 4) + j + ((lane >> 4) << 3);   // C/D layout: M=j (+8 hi half)
        float s = s2a[co], ad = a2a[co];
        float p = gelu_exact(acc0[j] * s + ad) + gelu_exact(acc1[j] * s + ad)
                + gelu_exact(acc2[j] * s + ad) + gelu_exact(acc3[j] * s + ad);
        p += __shfl_xor(p, 1, 32);
        p += __shfl_xor(p, 2, 32);
        p += __shfl_xor(p, 4, 32);
        p += __shfl_xor(p, 8, 32);
        if ((lane & 15) == 0)
            atomicAdd(&feats[((size_t)v * BB + b) * C2 + co], p);
    }
}

// ---------------------------------------------------------------------------
// Kernel 2: mean /512, MHA collapse (v-proj, out-proj), concat, final Linear.
// One block per batch row.
// ---------------------------------------------------------------------------
__global__ __launch_bounds__(128)
void tsrl_head(const float* __restrict__ feats,
               const float* __restrict__ in_w,  const float* __restrict__ in_b,
               const float* __restrict__ out_w, const float* __restrict__ out_b,
               const float* __restrict__ fin_w, const float* __restrict__ fin_b,
               float* __restrict__ out)
{
    __shared__ float bufA[NV * EMB];
    __shared__ float bufB[NV * EMB];
    const int b = blockIdx.x, tid = threadIdx.x;

    for (int i = tid; i < NV * EMB; i += 128) {
        int v = i >> 7, c = i & 127;
        bufA[i] = feats[((size_t)v * BB + b) * EMB + c] * (1.0f / 512.0f);
    }
    __syncthreads();
    // v = feats @ in_w[2E:].T + in_b[2E:]
    for (int o = tid; o < NV * EMB; o += 128) {
        int v = o >> 7, e = o & 127;
        const float* wr = in_w + (size_t)(2 * EMB + e) * EMB;
        const float* fr = bufA + v * EMB;
        float acc = in_b[2 * EMB + e];
        for (int c = 0; c < EMB; ++c) acc += fr[c] * wr[c];
        bufB[o] = acc;
    }
    __syncthreads();
    // attn = v @ out_w.T + out_b
    for (int o = tid; o < NV * EMB; o += 128) {
        int v = o >> 7, e = o & 127;
        const float* wr = out_w + (size_t)e * EMB;
        const float* fr = bufB + v * EMB;
        float acc = out_b[e];
        for (int c = 0; c < EMB; ++c) acc += fr[c] * wr[c];
        bufA[o] = acc;   // fused[b, v*E+e] ordering
    }
    __syncthreads();
    // out = fused @ fin_w.T + fin_b
    for (int p = tid; p < 35; p += 128) {
        const float* wr = fin_w + (size_t)p * (NV * EMB);
        float acc = fin_b[p];
        for (int i = 0; i < NV * EMB; ++i) acc += bufA[i] * wr[i];
        out[b * 35 + p] = acc;
    }
}

// ---------------------------------------------------------------------------
extern "C" void kernel_launch(void* const* d_in, const int* in_sizes, int n_in,
                              void* d_out, int out_size, void* d_ws, size_t ws_size,
                              hipStream_t stream) {
    (void)in_sizes; (void)n_in; (void)out_size; (void)ws_size;
    const float* x     = (const float*)d_in[0];
    const float* vc_w  = (const float*)d_in[1];
    const float* vc_b  = (const float*)d_in[2];
    const float* bn0_g = (const float*)d_in[3];
    const float* bn0_b = (const float*)d_in[4];
    const float* bn0_m = (const float*)d_in[5];
    const float* bn0_v = (const float*)d_in[6];
    const float* c1_w  = (const float*)d_in[7];
    const float* c1_b  = (const float*)d_in[8];
    const float* bn1_g = (const float*)d_in[9];
    const float* bn1_b = (const float*)d_in[10];
    const float* bn1_m = (const float*)d_in[11];
    const float* bn1_v = (const float*)d_in[12];
    const float* c2_w  = (const float*)d_in[13];
    const float* c2_b  = (const float*)d_in[14];
    const float* bn2_g = (const float*)d_in[15];
    const float* bn2_b = (const float*)d_in[16];
    const float* bn2_m = (const float*)d_in[17];
    const float* bn2_v = (const float*)d_in[18];
    const float* in_w  = (const float*)d_in[19];
    const float* in_b  = (const float*)d_in[20];
    const float* out_w = (const float*)d_in[21];
    const float* out_b = (const float*)d_in[22];
    const float* fin_w = (const float*)d_in[23];
    const float* fin_b = (const float*)d_in[24];
    float* out = (float*)d_out;
    float* feats = (float*)d_ws;                 // [NV][BB][C2] accumulator

    const int nfeat = NV * BB * C2;
    tsrl_zero<<<(nfeat + 255) / 256, 256, 0, stream>>>(feats, nfeat);

    const size_t shmem = (size_t)SMEM_FLOATS * sizeof(float);
    tsrl_fused<<<NV * BB * 8, 256, shmem, stream>>>(
        x, vc_w, vc_b, bn0_g, bn0_b, bn0_m, bn0_v,
        c1_w, c1_b, bn1_g, bn1_b, bn1_m, bn1_v,
        c2_w, c2_b, bn2_g, bn2_b, bn2_m, bn2_v, feats);

    tsrl_head<<<BB, 128, 0, stream>>>(feats, in_w, in_b, out_w, out_b,
                                      fin_w, fin_b, out);
}